// CoPooling_66228395704951
// MI455X (gfx1250) — compile-verified
//
#include <hip/hip_runtime.h>
#include <hip/hip_bf16.h>

// ---------------- problem constants ----------------
#define N_      6144
#define E_      98304
#define NHID_   64
#define DOUT_   32
#define KHOPS_  10
#define KKEEP_  4916
#define COALN_  (E_ + N_)          // 104448 values for percentile
#define SORTN_  8192               // pow2 >= N_

typedef float v2f __attribute__((ext_vector_type(2)));
typedef float v8f __attribute__((ext_vector_type(8)));

static __host__ __device__ inline int cdiv(int a, int b) { return (a + b - 1) / b; }

// ---------------- generic fill ----------------
__global__ void cp_fill(float* __restrict__ p, size_t n, float v) {
    size_t i = (size_t)blockIdx.x * blockDim.x + threadIdx.x;
    if (i < n) p[i] = v;
}

// ---------------- degree / dinv (gcn_norm with self loops) ----------------
__global__ void cp_deg(const int* __restrict__ c0, float* __restrict__ deg) {
    int i = blockIdx.x * blockDim.x + threadIdx.x;
    if (i < E_) atomicAdd(&deg[c0[i]], 1.0f);
}
__global__ void cp_dinv(float* __restrict__ deg) {
    int i = blockIdx.x * blockDim.x + threadIdx.x;
    if (i < N_) { float d = deg[i]; deg[i] = (d > 0.0f) ? (1.0f / sqrtf(d)) : 0.0f; }
}

// ---------------- GPR propagation hops ----------------
__global__ void cp_init_h(const float* __restrict__ x, const float* __restrict__ temp,
                          float* __restrict__ h) {
    int i = blockIdx.x * blockDim.x + threadIdx.x;
    if (i < N_ * NHID_) h[i] = temp[0] * x[i];
}
// xnext[i,:] = dinv[i]^2 * xk[i,:]   (self-loop edge)
__global__ void cp_hop_init(const float* __restrict__ xk, const float* __restrict__ dinv,
                            float* __restrict__ xnext) {
    int idx = blockIdx.x * blockDim.x + threadIdx.x;
    if (idx < N_ * NHID_) {
        int i = idx >> 6;
        float di = dinv[i];
        xnext[idx] = di * di * xk[idx];
    }
}
// scatter: xnext[c,:] += dinv[r]*dinv[c] * xk[r,:]
__global__ void cp_hop_edges(const int* __restrict__ r0, const int* __restrict__ c0,
                             const float* __restrict__ dinv, const float* __restrict__ xk,
                             float* __restrict__ xnext) {
    int idx = blockIdx.x * blockDim.x + threadIdx.x;
    if (idx >= E_ * NHID_) return;
    int e = idx >> 6, hh = idx & (NHID_ - 1);
    int r = r0[e], c = c0[e];
    float nrm = dinv[r] * dinv[c];
    atomicAdd(&xnext[c * NHID_ + hh], nrm * xk[r * NHID_ + hh]);
}
__global__ void cp_hop_accum(float* __restrict__ h, const float* __restrict__ xnext,
                             const float* __restrict__ temp, int kidx) {
    int i = blockIdx.x * blockDim.x + threadIdx.x;
    if (i < N_ * NHID_) h[i] += temp[kidx] * xnext[i];
}

// ---------------- attention scores ----------------
// wsrc[h] = sum_d W_proj[d,h]*a_src[d]  (fold projection + attention vectors)
__global__ void cp_make_wvec(const float* __restrict__ Wp, const float* __restrict__ a_src,
                             const float* __restrict__ a_tgt, float* __restrict__ wsrc,
                             float* __restrict__ wtgt) {
    int hh = threadIdx.x;  // 64
    float s1 = 0.f, s2 = 0.f;
    for (int d = 0; d < DOUT_; ++d) {
        float w = Wp[d * NHID_ + hh];
        s1 += w * a_src[d];
        s2 += w * a_tgt[d];
    }
    wsrc[hh] = s1; wtgt[hh] = s2;
}
__global__ void cp_node_scores(const float* __restrict__ h, const float* __restrict__ wsrc,
                               const float* __restrict__ wtgt, float* __restrict__ s_src,
                               float* __restrict__ s_tgt) {
    __shared__ float red[2 * NHID_];
    int i = blockIdx.x, t = threadIdx.x;  // 64 threads
    float v = h[i * NHID_ + t];
    red[t] = v * wsrc[t];
    red[NHID_ + t] = v * wtgt[t];
    __syncthreads();
    for (int s = 32; s > 0; s >>= 1) {
        if (t < s) { red[t] += red[t + s]; red[NHID_ + t] += red[NHID_ + t + s]; }
        __syncthreads();
    }
    if (t == 0) { s_src[i] = red[0]; s_tgt[i] = red[NHID_]; }
}
__global__ void cp_att_edge(const int* __restrict__ r0, const int* __restrict__ c0,
                            const float* __restrict__ s_src, const float* __restrict__ s_tgt,
                            float* __restrict__ att_e) {
    int e = blockIdx.x * blockDim.x + threadIdx.x;
    if (e < E_) {
        float z = s_src[r0[e]] + s_tgt[c0[e]];
        att_e[e] = 1.0f / (1.0f + expf(-z));
    }
}

// ---------------- dense A / Cnt build ----------------
__global__ void cp_scatter(const int* __restrict__ r0, const int* __restrict__ c0,
                           const float* __restrict__ att_e, float* __restrict__ A,
                           float* __restrict__ Cnt) {
    int e = blockIdx.x * blockDim.x + threadIdx.x;
    if (e < E_) {
        size_t idx = (size_t)r0[e] * N_ + c0[e];
        atomicAdd(&A[idx], att_e[e]);
        atomicAdd(&Cnt[idx], 1.0f);
    }
}
__global__ void cp_scatter_diag(float* __restrict__ A, float* __restrict__ Cnt) {
    int i = blockIdx.x * blockDim.x + threadIdx.x;
    if (i < N_) {
        size_t idx = (size_t)i * N_ + i;
        atomicAdd(&A[idx], 1.0f);
        atomicAdd(&Cnt[idx], 1.0f);
    }
}
// M = where(Csym>0, Asym/max(Csym,1), 0), written in place into A (symmetric pairs)
__global__ void cp_sym_mean(float* __restrict__ A, const float* __restrict__ Cnt) {
    size_t idx = (size_t)blockIdx.x * blockDim.x + threadIdx.x;
    if (idx >= (size_t)N_ * N_) return;
    int i = (int)(idx / N_), j = (int)(idx % N_);
    if (i > j) return;
    size_t ij = (size_t)i * N_ + j, ji = (size_t)j * N_ + i;
    float cs = Cnt[ij] + Cnt[ji];
    float m = 0.0f;
    if (cs > 0.0f) m = (A[ij] + A[ji]) / fmaxf(cs, 1.0f);
    A[ij] = m;
    if (ji != ij) A[ji] = m;
}
// coal[e] = M[rs[e], cs[e]] over original edges + self loops
__global__ void cp_coal(const int* __restrict__ r0, const int* __restrict__ c0,
                        const float* __restrict__ M, float* __restrict__ coal) {
    int e = blockIdx.x * blockDim.x + threadIdx.x;
    if (e >= COALN_) return;
    int r = (e < E_) ? r0[e] : (e - E_);
    int c = (e < E_) ? c0[e] : (e - E_);
    coal[e] = M[(size_t)r * N_ + c];
}

// ---------------- exact k-th smallest via 4-pass radix select ----------------
// state: st[0]=prefix key, st[1]=remaining rank, st[2..257]=hist
__global__ void cp_sel_init(unsigned* __restrict__ st, unsigned rank0) {
    int t = threadIdx.x;
    if (t == 0) { st[0] = 0u; st[1] = rank0; }
    if (t < 256) st[2 + t] = 0u;
}
__device__ __forceinline__ unsigned f2key(float f) {
    unsigned u = __float_as_uint(f);
    return (u & 0x80000000u) ? ~u : (u ^ 0x80000000u);
}
__global__ void cp_sel_count(const float* __restrict__ vals, int n,
                             unsigned* __restrict__ st, int shift) {
    int i = blockIdx.x * blockDim.x + threadIdx.x;
    if (i >= n) return;
    unsigned key = f2key(vals[i]);
    unsigned prefix = st[0];
    if (shift < 24) {
        unsigned mask = 0xFFFFFFFFu << (shift + 8);
        if ((key & mask) != (prefix & mask)) return;
    }
    atomicAdd(&st[2 + ((key >> shift) & 0xFFu)], 1u);
}
__global__ void cp_sel_pick(unsigned* __restrict__ st, int shift,
                            float* __restrict__ outval, int final_pass) {
    unsigned rank = st[1];
    unsigned cum = 0, b = 255;
    for (unsigned bb = 0; bb < 256; ++bb) {
        unsigned c = st[2 + bb];
        if (cum + c > rank) { b = bb; break; }
        cum += c;
    }
    st[0] |= (b << shift);
    st[1] = rank - cum;
    for (int j = 0; j < 256; ++j) st[2 + j] = 0u;
    if (final_pass) {
        unsigned key = st[0];
        unsigned u = (key & 0x80000000u) ? (key ^ 0x80000000u) : ~key;
        *outval = __uint_as_float(u);
    }
}
__global__ void cp_cut_combine(const float* __restrict__ lo, const float* __restrict__ hi,
                               float* __restrict__ cut) {
    // percentile position = 0.4*(COALN_-1) = 41778.8 -> lerp weight 0.8
    *cut = lo[0] + 0.8f * (hi[0] - lo[0]);
}

// ---------------- threshold + NodeInformationScore pieces ----------------
__global__ void cp_thresh(float* __restrict__ M, const float* __restrict__ cut) {
    size_t idx = (size_t)blockIdx.x * blockDim.x + threadIdx.x;
    if (idx >= (size_t)N_ * N_) return;
    float v = M[idx];
    M[idx] = (v >= *cut) ? v : 0.0f;
}
__global__ void cp_row_dinv(const float* __restrict__ Ad, float* __restrict__ dinvc) {
    __shared__ float red[256];
    int i = blockIdx.x, t = threadIdx.x;
    float s = 0.f;
    for (int j = t; j < N_; j += 256) s += Ad[(size_t)i * N_ + j];
    red[t] = s; __syncthreads();
    for (int st = 128; st > 0; st >>= 1) {
        if (t < st) red[t] += red[t + st];
        __syncthreads();
    }
    if (t == 0) { float d = red[0]; dinvc[i] = (d > 0.0f) ? (1.0f / sqrtf(d)) : 0.0f; }
}
__global__ void cp_scale_rows(const float* __restrict__ x, const float* __restrict__ dinvc,
                              float* __restrict__ xs) {
    int idx = blockIdx.x * blockDim.x + threadIdx.x;
    if (idx < N_ * NHID_) xs[idx] = dinvc[idx >> 6] * x[idx];
}
__global__ void cp_score(const float* __restrict__ x, const float* __restrict__ y,
                         const float* __restrict__ dinvc, float* __restrict__ score) {
    __shared__ float red[NHID_];
    int i = blockIdx.x, t = threadIdx.x;
    float v = fabsf(x[i * NHID_ + t] - dinvc[i] * y[i * NHID_ + t]);
    red[t] = v; __syncthreads();
    for (int s = 32; s > 0; s >>= 1) {
        if (t < s) red[t] += red[t + s];
        __syncthreads();
    }
    if (t == 0) score[i] = red[0];
}

// ---------------- WMMA f32 GEMM: C[M x 64] = A[rows x N_] @ B[N_ x 64] ----------------
// 128 threads = 4 waves; each wave owns a 16x16 tile of C (cols wave*16..+15).
// A fragment (16x4 f32): lane L: m=L&15, kb=(L>>4)*2 -> {A[m,kb], A[m,kb+1]}
// B fragment (4x16 f32): lane L: n=L&15, kb=(L>>4)*2 -> {B[kb,n], B[kb+1,n]}
// C (16x16 f32, 8 VGPRs): acc[r] at lane L -> (M = r + (L>=16?8:0), N = L&15)
__global__ __launch_bounds__(128) void cp_gemm_wmma(
        const float* __restrict__ A, const int* __restrict__ rowidx, int Mrows,
        const float* __restrict__ B, float* __restrict__ C) {
    __shared__ float As[16 * 64];
    __shared__ float Bs[64 * 64];
    const int block_row = blockIdx.x * 16;
    const int tid = threadIdx.x;
    const int wave = tid >> 5;
    const int lane = tid & 31;
    v8f acc = {};

    for (int k0 = 0; k0 < N_; k0 += 64) {
        // stage A tile (16 x 64), coalesced row segments
        for (int i = tid; i < 16 * 64; i += 128) {
            int r = i >> 6, kk = i & 63;
            int gr = block_row + r;
            float v = 0.0f;
            if (gr < Mrows) {
                int ar = rowidx ? rowidx[gr] : gr;
                const float* src = &A[(size_t)ar * N_ + k0 + kk];
                v = *src;
                if (k0 + 64 < N_) __builtin_prefetch(src + 64, 0, 1);  // global_prefetch_b8
            }
            As[i] = v;
        }
        // stage B tile (64 x 64)
        for (int i = tid; i < 64 * 64; i += 128) {
            int r = i >> 6, nn = i & 63;
            Bs[i] = B[(size_t)(k0 + r) * NHID_ + nn];
        }
        __syncthreads();

        const int ncol0 = wave * 16;
        const int m = lane & 15;
        const int n = lane & 15;
        const int kb = (lane >> 4) * 2;
        #pragma unroll
        for (int kk = 0; kk < 64; kk += 4) {
            v2f afrag, bfrag;
            afrag.x = As[m * 64 + kk + kb];
            afrag.y = As[m * 64 + kk + kb + 1];
            bfrag.x = Bs[(kk + kb) * 64 + ncol0 + n];
            bfrag.y = Bs[(kk + kb + 1) * 64 + ncol0 + n];
            acc = __builtin_amdgcn_wmma_f32_16x16x4_f32(
                false, afrag, false, bfrag, (short)0, acc, false, false);
        }
        __syncthreads();
    }

    const int n = lane & 15;
    const int mofs = (lane >> 4) * 8;
    const int ncol0 = wave * 16;
    #pragma unroll
    for (int r = 0; r < 8; ++r) {
        int gm = block_row + mofs + r;
        if (gm < Mrows) C[(size_t)gm * NHID_ + ncol0 + n] = acc[r];
    }
}

// ---------------- full descending bitonic sort in LDS -> perm ----------------
__global__ __launch_bounds__(1024) void cp_topk(const float* __restrict__ score,
                                                int* __restrict__ perm) {
    __shared__ unsigned long long keys[SORTN_];  // 64 KB of the 320 KB/WGP LDS
    int tid = threadIdx.x;
    for (int i = tid; i < SORTN_; i += 1024) {
        unsigned long long k = 0ull;
        if (i < N_) {
            unsigned fb = __float_as_uint(score[i]);  // scores are >= 0
            k = ((unsigned long long)fb << 32) | (unsigned long long)(0xFFFFFFFFu - (unsigned)i);
        }
        keys[i] = k;
    }
    __syncthreads();
    for (int size = 2; size <= SORTN_; size <<= 1) {
        for (int stride = size >> 1; stride > 0; stride >>= 1) {
            for (int t = tid; t < SORTN_ / 2; t += 1024) {
                int lo = 2 * t - (t & (stride - 1));
                int hi = lo + stride;
                bool desc = ((lo & size) == 0);
                unsigned long long a = keys[lo], b = keys[hi];
                bool sw = desc ? (a < b) : (a > b);
                if (sw) { keys[lo] = b; keys[hi] = a; }
            }
            __syncthreads();
        }
    }
    for (int i = tid; i < KKEEP_; i += 1024)
        perm[i] = (int)(0xFFFFFFFFu - (unsigned)(keys[i] & 0xFFFFFFFFull));
}

// ---------------- final MLP: relu([x[perm], agg] @ W + b) ----------------
__global__ void cp_mlp(const float* __restrict__ x, const int* __restrict__ perm,
                       const float* __restrict__ agg, const float* __restrict__ W,
                       const float* __restrict__ bias, float* __restrict__ out) {
    __shared__ float xa[2 * NHID_];
    int i = blockIdx.x;       // < KKEEP_
    int j = threadIdx.x;      // 64
    int p = perm[i];
    xa[j] = x[(size_t)p * NHID_ + j];
    xa[NHID_ + j] = agg[(size_t)i * NHID_ + j];
    __syncthreads();
    float acc = bias[j];
    #pragma unroll 8
    for (int k = 0; k < 2 * NHID_; ++k) acc += xa[k] * W[k * NHID_ + j];
    out[(size_t)i * NHID_ + j] = fmaxf(acc, 0.0f);
}

// ================= host orchestration =================
extern "C" void kernel_launch(void* const* d_in, const int* in_sizes, int n_in,
                              void* d_out, int out_size, void* d_ws, size_t ws_size,
                              hipStream_t stream) {
    const float* x      = (const float*)d_in[0];
    const int*   eidx   = (const int*)d_in[1];
    const int*   r0     = eidx;
    const int*   c0     = eidx + E_;
    const float* temp   = (const float*)d_in[4];
    const float* Wp     = (const float*)d_in[5];
    const float* a_src  = (const float*)d_in[6];
    const float* a_tgt  = (const float*)d_in[7];
    const float* weight = (const float*)d_in[8];
    const float* bias   = (const float*)d_in[9];
    float* out = (float*)d_out;

    const size_t NN = (size_t)N_ * N_;
    float* ws = (float*)d_ws;
    size_t o = 0;
    float* A     = ws + o; o += NN;          // A -> M -> att_dense (in place)
    float* Cnt   = ws + o; o += NN;
    float* bx0   = ws + o; o += (size_t)N_ * NHID_;
    float* bx1   = ws + o; o += (size_t)N_ * NHID_;
    float* hbuf  = ws + o; o += (size_t)N_ * NHID_;
    float* ybuf  = ws + o; o += (size_t)N_ * NHID_;
    float* xsbuf = ws + o; o += (size_t)N_ * NHID_;
    float* dinv  = ws + o; o += N_;          // deg -> dinv in place
    float* dinvc = ws + o; o += N_;
    float* ssrc  = ws + o; o += N_;
    float* stgt  = ws + o; o += N_;
    float* score = ws + o; o += N_;
    float* atte  = ws + o; o += E_;
    float* coal  = ws + o; o += COALN_;
    float* wsrc  = ws + o; o += NHID_;
    float* wtgt  = ws + o; o += NHID_;
    float* cutb  = ws + o; o += 4;           // [lo, hi, cut]
    unsigned* selst = (unsigned*)(ws + o); o += 264;
    int* perm    = (int*)(ws + o); o += N_;
    float* agg   = ws + o; o += (size_t)KKEEP_ * NHID_;
    (void)ws_size; (void)n_in; (void)in_sizes; (void)out_size;

    const int T = 256;
    const int gNN  = (int)((NN + T - 1) / T);
    const int gNH  = cdiv(N_ * NHID_, T);
    const int gE   = cdiv(E_, T);
    const int gEH  = cdiv(E_ * NHID_, T);
    const int gN   = cdiv(N_, T);
    const int gCO  = cdiv(COALN_, T);

    // 1) degrees + dinv (self loops contribute the initial 1.0)
    cp_fill<<<gNN, T, 0, stream>>>(A, NN, 0.0f);
    cp_fill<<<gNN, T, 0, stream>>>(Cnt, NN, 0.0f);
    cp_fill<<<gN, T, 0, stream>>>(dinv, (size_t)N_, 1.0f);
    cp_deg<<<gE, T, 0, stream>>>(c0, dinv);
    cp_dinv<<<gN, T, 0, stream>>>(dinv);

    // 2) GPR propagation: h = sum_k temp[k] * x_k
    cp_init_h<<<gNH, T, 0, stream>>>(x, temp, hbuf);
    const float* cur = x;
    float* nxt = bx0;
    float* other = bx1;
    for (int k = 0; k < KHOPS_; ++k) {
        cp_hop_init<<<gNH, T, 0, stream>>>(cur, dinv, nxt);
        cp_hop_edges<<<gEH, T, 0, stream>>>(r0, c0, dinv, cur, nxt);
        cp_hop_accum<<<gNH, T, 0, stream>>>(hbuf, nxt, temp, k + 1);
        cur = nxt;
        nxt = other;
        other = (float*)cur == bx0 ? bx1 : bx0;
        other = (nxt == bx0) ? bx1 : bx0;
    }

    // 3) attention scores per edge
    cp_make_wvec<<<1, NHID_, 0, stream>>>(Wp, a_src, a_tgt, wsrc, wtgt);
    cp_node_scores<<<N_, NHID_, 0, stream>>>(hbuf, wsrc, wtgt, ssrc, stgt);
    cp_att_edge<<<gE, T, 0, stream>>>(r0, c0, ssrc, stgt, atte);

    // 4) dense A/Cnt, symmetrized mean M (in place in A)
    cp_scatter<<<gE, T, 0, stream>>>(r0, c0, atte, A, Cnt);
    cp_scatter_diag<<<gN, T, 0, stream>>>(A, Cnt);
    cp_sym_mean<<<gNN, T, 0, stream>>>(A, Cnt);
    cp_coal<<<gCO, T, 0, stream>>>(r0, c0, A, coal);

    // 5) exact 40th percentile (linear interpolation) via radix select
    const unsigned K0 = 41778u, K1 = 41779u;  // floor/ceil of 0.4*(COALN_-1)
    for (int which = 0; which < 2; ++which) {
        cp_sel_init<<<1, 256, 0, stream>>>(selst, which == 0 ? K0 : K1);
        for (int shift = 24; shift >= 0; shift -= 8) {
            cp_sel_count<<<gCO, T, 0, stream>>>(coal, COALN_, selst, shift);
            cp_sel_pick<<<1, 1, 0, stream>>>(selst, shift, cutb + which, shift == 0 ? 1 : 0);
        }
    }
    cp_cut_combine<<<1, 1, 0, stream>>>(cutb, cutb + 1, cutb + 2);

    // 6) threshold -> att_dense (in place), degrees, scaled features
    cp_thresh<<<gNN, T, 0, stream>>>(A, cutb + 2);
    cp_row_dinv<<<N_, 256, 0, stream>>>(A, dinvc);
    cp_scale_rows<<<gNH, T, 0, stream>>>(x, dinvc, xsbuf);

    // 7) y = att_dense @ xs  (WMMA f32) ; score_i = sum_h |x - dinvc*y|
    cp_gemm_wmma<<<N_ / 16, 128, 0, stream>>>(A, nullptr, N_, xsbuf, ybuf);
    cp_score<<<N_, NHID_, 0, stream>>>(x, ybuf, dinvc, score);

    // 8) exact descending top-k with index tiebreak (full bitonic sort in LDS)
    cp_topk<<<1, 1024, 0, stream>>>(score, perm);

    // 9) agg = att_dense[perm] @ x  (WMMA f32, gathered rows)
    cp_gemm_wmma<<<cdiv(KKEEP_, 16), 128, 0, stream>>>(A, perm, KKEEP_, x, agg);

    // 10) out = relu([x[perm], agg] @ W + b)
    cp_mlp<<<KKEEP_, NHID_, 0, stream>>>(x, perm, agg, weight, bias, out);
}